// Eikonal2D_14061722927718
// MI455X (gfx1250) — compile-verified
//
#include <hip/hip_runtime.h>
#include <hip/hip_bf16.h>
#include <stdint.h>

#define NXg   128
#define NYg   128
#define NCELL (NXg * NYg)          // 16384 cells, 64 KB f32 per buffer
#define NITER 320                  // even -> final field ends in buf[0]
#define NSTA  24
#define NPICK 200000
#define INFV  1.0e6f
#define TBLK  1024                 // 32 wave32 waves -> fills one WGP
#define CPT   (NCELL / TBLK)       // 16 cells per thread

// ---------------------------------------------------------------------------
// Kernel 1: eikonal solve, one workgroup per (phase, station).
// Double-buffered grid in LDS (128 KB of the WGP's 320 KB), center value
// carried in VGPRs, one barrier per Jacobi sweep. Velocity staged in with
// CDNA5 async global->LDS loads; result streamed out with async store-from-LDS.
// ---------------------------------------------------------------------------
__global__ __launch_bounds__(TBLK) void eikonal_kernel(
    const float* __restrict__ vp, const float* __restrict__ vs,
    const float* __restrict__ station_loc, float* __restrict__ T)
{
    __shared__ float u[2][NCELL];  // buf0 at LDS offset 0 (only shared array)
    const int b   = blockIdx.x;    // 0..47
    const int ph  = b / NSTA;      // 0 = P, 1 = S
    const int st  = b - ph * NSTA;
    const int tid = threadIdx.x;

    // station grid index (jnp.round == round-to-nearest-even == rintf)
    int ixs = (int)rintf(station_loc[st * 3 + 0]);
    int iys = (int)rintf(station_loc[st * 3 + 1]);
    ixs = min(max(ixs, 0), NXg - 1);
    iys = min(max(iys, 0), NYg - 1);
    const int src = ixs * NYg + iys;

    const float* vel = (ph == 0) ? vp : vs;

    // --- CDNA5 async global -> LDS stage of the velocity grid (into buf0) --
    #pragma unroll
    for (int k = 0; k < CPT; ++k) {
        unsigned idx = (unsigned)(tid + k * TBLK);
        unsigned off = idx * 4u;                 // byte offset (LDS == global)
        asm volatile("global_load_async_to_lds_b32 %0, %1, %2"
                     :: "v"(off), "v"(off), "s"(vel) : "memory");
    }
    asm volatile("s_wait_asynccnt 0" ::: "memory");

    // each thread reads back exactly the cells it loaded -> no barrier needed
    float fh[CPT];
    #pragma unroll
    for (int k = 0; k < CPT; ++k)
        fh[k] = 1.0f / u[0][tid + k * TBLK];     // fh = (1/v) * H, H = 1

    __syncthreads();                             // all async writes consumed
    float uc[CPT];                               // register-carried center
    #pragma unroll
    for (int k = 0; k < CPT; ++k) {
        int idx = tid + k * TBLK;
        uc[k] = (idx == src) ? 0.0f : INFV;
        u[0][idx] = uc[k];
    }
    __syncthreads();

    // --- 320 Jacobi sweeps: read neighbors from buf[p], write buf[p^1] -----
    int p = 0;
    for (int it = 0; it < NITER; ++it) {
        const float* __restrict__ ub = u[p];
        float*       __restrict__ un = u[p ^ 1];
        #pragma unroll
        for (int k = 0; k < CPT; ++k) {
            int idx = tid + k * TBLK;
            int ix  = idx >> 7;
            int iy  = idx & (NYg - 1);
            float xm = (ix > 0)       ? ub[idx - NYg] : INFV;
            float xp = (ix < NXg - 1) ? ub[idx + NYg] : INFV;
            float ym = (iy > 0)       ? ub[idx - 1]   : INFV;
            float yp = (iy < NYg - 1) ? ub[idx + 1]   : INFV;
            float a  = fminf(xm, xp);
            float bb = fminf(ym, yp);
            float d  = fabsf(a - bb);
            float f  = fh[k];
            float one_sided = fminf(a, bb) + f;
            float two_sided = 0.5f * (a + bb +
                __builtin_amdgcn_sqrtf(fmaxf(2.0f * f * f - d * d, 0.0f)));
            float cand = (d >= f) ? one_sided : two_sided;
            uc[k] = fminf(uc[k], cand);          // src stays 0 automatically
            un[idx] = uc[k];
        }
        p ^= 1;
        __syncthreads();
    }
    // NITER even -> p == 0 and the freshest field lives in buf0 (LDS offset 0)

    // --- CDNA5 async LDS -> global stream-out of the travel-time tile ------
    float* Tout = T + (size_t)b * NCELL;
    #pragma unroll
    for (int k = 0; k < CPT; ++k) {
        unsigned idx = (unsigned)(tid + k * TBLK);
        unsigned off = idx * 4u;
        asm volatile("global_store_async_from_lds_b32 %0, %1, %2"
                     :: "v"(off), "v"(off), "s"(Tout) : "memory");
    }
    asm volatile("s_wait_asynccnt 0" ::: "memory");
}

// ---------------------------------------------------------------------------
// Kernel 2: per-pick bilinear interpolation + predicted arrival time
// ---------------------------------------------------------------------------
__global__ __launch_bounds__(256) void interp_kernel(
    const float* __restrict__ T,
    const float* __restrict__ event_loc, const float* __restrict__ event_time,
    const int* __restrict__ pick_station, const int* __restrict__ pick_phase,
    const int* __restrict__ pick_event, float* __restrict__ at_out)
{
    int i = blockIdx.x * blockDim.x + threadIdx.x;
    if (i >= NPICK) return;
    int st = pick_station[i], ph = pick_phase[i], ev = pick_event[i];
    float x = event_loc[ev * 3 + 0];
    float y = event_loc[ev * 3 + 1];
    int ir0 = (int)floorf(x); ir0 = min(max(ir0, 0), NXg - 2);
    int iz0 = (int)floorf(y); iz0 = min(max(iz0, 0), NYg - 2);
    float xc = fminf(fmaxf(x, 0.0f), (float)(NXg - 1));
    float yc = fminf(fmaxf(y, 0.0f), (float)(NYg - 1));
    float r0 = (float)ir0, z0 = (float)iz0;
    const float* Tt = T + ((size_t)(ph * NSTA + st) << 14) + (ir0 << 7) + iz0;
    float Q00 = Tt[0], Q01 = Tt[1], Q10 = Tt[NYg], Q11 = Tt[NYg + 1];
    float wx0 = xc - r0, wx1 = (r0 + 1.0f) - xc;
    float wy0 = yc - z0, wy1 = (z0 + 1.0f) - yc;
    float tt = Q00 * wx1 * wy1 + Q10 * wx0 * wy1
             + Q01 * wx1 * wy0 + Q11 * wx0 * wy0;   // /(H*H) == /1
    at_out[i] = event_time[ev] + tt;
}

// ---------------------------------------------------------------------------
// Kernel 3: deterministic per-group SSE / count (48 groups, fixed-tree reduce)
// ---------------------------------------------------------------------------
__global__ __launch_bounds__(256) void segsum_kernel(
    const float* __restrict__ at, const float* __restrict__ phase_time,
    const int* __restrict__ pick_station, const int* __restrict__ pick_phase,
    float* __restrict__ sse, float* __restrict__ cnt)
{
    __shared__ float ss[256];
    __shared__ float sc[256];
    const int g   = blockIdx.x;      // 0..47
    const int tid = threadIdx.x;
    float s = 0.0f, c = 0.0f;
    for (int i = tid; i < NPICK; i += 256) {
        int gid = pick_station[i] * 2 + pick_phase[i];
        if (gid == g) {
            float r = at[i] - phase_time[i];
            s += r * r;
            c += 1.0f;
        }
    }
    ss[tid] = s; sc[tid] = c;
    __syncthreads();
    for (int off = 128; off > 0; off >>= 1) {
        if (tid < off) { ss[tid] += ss[tid + off]; sc[tid] += sc[tid + off]; }
        __syncthreads();
    }
    if (tid == 0) { sse[g] = ss[0]; cnt[g] = sc[0]; }
}

// ---------------------------------------------------------------------------
// Kernel 4: loss = sum_g sse[g]/max(cnt[g],1), reduced with an exact-f32
// V_WMMA_F32_16X16X4_F32: 48 ratios in the 16x4 A tile, B = ones(4x16),
// so each D row is the exact K-sum; 8 VGPR adds + SWAPX16 swizzle finish.
// ---------------------------------------------------------------------------
typedef __attribute__((ext_vector_type(2))) float v2f;
typedef __attribute__((ext_vector_type(8))) float v8f;

__global__ __launch_bounds__(32) void loss_kernel(
    const float* __restrict__ sse, const float* __restrict__ cnt,
    float* __restrict__ loss_out)
{
    const int lane = threadIdx.x;                 // one wave32
    // A-tile slot (lane, reg r) holds value index lane + 32*r (48 real, 16 zero)
    float r0 = sse[lane] / fmaxf(cnt[lane], 1.0f);       // lane in 0..31 < 48
    float r1 = 0.0f;
    if (lane < 16) {
        int j = lane + 32;                                // 32..47
        r1 = sse[j] / fmaxf(cnt[j], 1.0f);
    }
    v2f a;    a.x = r0;  a.y = r1;
    v2f ones; ones.x = 1.0f; ones.y = 1.0f;
    v8f cacc = {};
    // 8 args: (neg_a, A, neg_b, B, c_mod, C, reuse_a, reuse_b)
    v8f d = __builtin_amdgcn_wmma_f32_16x16x4_f32(
        false, a, false, ones, (short)0, cacc, false, false);
    float s = d.s0 + d.s1 + d.s2 + d.s3 + d.s4 + d.s5 + d.s6 + d.s7;
    // lanes 0-15 hold sum of rows 0..7; lanes 16-31 rows 8..15 -> SWAPX16 add
    int sw = __builtin_amdgcn_ds_swizzle(__float_as_int(s), 0x401F);
    float tot = s + __int_as_float(sw);
    if (lane == 0) loss_out[0] = tot;
}

// ---------------------------------------------------------------------------
extern "C" void kernel_launch(void* const* d_in, const int* in_sizes, int n_in,
                              void* d_out, int out_size, void* d_ws, size_t ws_size,
                              hipStream_t stream) {
    (void)in_sizes; (void)n_in; (void)out_size; (void)ws_size;
    const float* vp           = (const float*)d_in[0];
    const float* vs           = (const float*)d_in[1];
    const float* station_loc  = (const float*)d_in[2];
    const float* event_loc    = (const float*)d_in[3];
    const float* event_time   = (const float*)d_in[4];
    const int*   pick_station = (const int*)d_in[5];
    const int*   pick_phase   = (const int*)d_in[6];
    const int*   pick_event   = (const int*)d_in[7];
    const float* phase_time   = (const float*)d_in[8];

    float* out = (float*)d_out;                 // [0..NPICK-1]=at, [NPICK]=loss
    char*  ws  = (char*)d_ws;
    float* T   = (float*)ws;                    // 48 * 16384 f32 = 3 MB
    float* sse = (float*)(ws + (size_t)48 * NCELL * sizeof(float));
    float* cnt = sse + 48;

    eikonal_kernel<<<48, TBLK, 0, stream>>>(vp, vs, station_loc, T);
    interp_kernel<<<(NPICK + 255) / 256, 256, 0, stream>>>(
        T, event_loc, event_time, pick_station, pick_phase, pick_event, out);
    segsum_kernel<<<48, 256, 0, stream>>>(
        out, phase_time, pick_station, pick_phase, sse, cnt);
    loss_kernel<<<1, 32, 0, stream>>>(sse, cnt, out + NPICK);
}